// VQVariationalAutoencoder_3504693314186
// MI455X (gfx1250) — compile-verified
//
#include <hip/hip_runtime.h>

typedef __attribute__((ext_vector_type(2))) float v2f;
typedef __attribute__((ext_vector_type(8))) float v8f;

#define WMMA_F32(a, b, c) \
  __builtin_amdgcn_wmma_f32_16x16x4_f32(false, (a), false, (b), (short)0, (c), false, false)

// ---------------------------------------------------------------------------
// CDNA5 async global->LDS staging (ASYNCcnt path). Guarded so the file
// compiles on toolchains that do not expose the builtins.
// Probe-derived signatures (from round-2 diagnostics):
//   b32 : (as1 int*, as3 int*, imm offset, imm cpol)
//   b128: (as1 v4i*, as3 v4i*, imm offset, imm cpol)   [v4i = vector_size(16)]
// ---------------------------------------------------------------------------
#if __has_builtin(__builtin_amdgcn_global_load_async_to_lds_b32) && \
    __has_builtin(__builtin_amdgcn_global_load_async_to_lds_b128)
#define HAS_ASYNC_LDS 1
#else
#define HAS_ASYNC_LDS 0
#endif

#if HAS_ASYNC_LDS
typedef int v4i_gcc __attribute__((__vector_size__(16)));
typedef __attribute__((address_space(1))) int*     gptr_b32;
typedef __attribute__((address_space(3))) int*     lptr_b32;
typedef __attribute__((address_space(1))) v4i_gcc* gptr_b128;
typedef __attribute__((address_space(3))) v4i_gcc* lptr_b128;

__device__ __forceinline__ void async_g2l_b32(const float* g, float* l) {
  __builtin_amdgcn_global_load_async_to_lds_b32(
      (gptr_b32)(unsigned long long)(const void*)g,
      (lptr_b32)(unsigned)(unsigned long long)(void*)l, 0, 0);
}
__device__ __forceinline__ void async_g2l_b128(const float* g, float* l) {
  __builtin_amdgcn_global_load_async_to_lds_b128(
      (gptr_b128)(unsigned long long)(const void*)g,
      (lptr_b128)(unsigned)(unsigned long long)(void*)l, 0, 0);
}
__device__ __forceinline__ void wait_async0() {
#if __has_builtin(__builtin_amdgcn_s_wait_asynccnt)
  __builtin_amdgcn_s_wait_asynccnt(0);
#else
  asm volatile("s_wait_asynccnt 0" ::: "memory");
#endif
}
#endif

// ---------------------------------------------------------------------------
// Sizes
// ---------------------------------------------------------------------------
#define Sdim 576
#define Fdim 700
#define LAT 64
#define KCODE 1024
#define Bdim 16
#define Mrows (Bdim * Sdim)          // 9216
#define SF (Sdim * Fdim)             // 403200
#define SD (Sdim * LAT)              // 36864

// ---------------------------------------------------------------------------
// Generic fp32 WMMA GEMM:  C = act(A[M,K] @ B[K,N] + bias), row-major.
// Block: 256 threads = 8 waves (2 in M x 4 in N). Block tile 64x128, BK=16.
// Each wave: 32x32 via 2x2 v_wmma_f32_16x16x4_f32 accumulators.
// Double-buffered LDS pipeline; stage s+1 loads overlap stage s WMMAs.
// ---------------------------------------------------------------------------
template <int ACT>   // 0 = none, 1 = relu
__global__ __launch_bounds__(256) void gemm_wmma(
    const float* __restrict__ A, const float* __restrict__ B,
    const float* __restrict__ bias, float* __restrict__ C,
    int M, int N, int K) {
  __shared__ float Als[2][64 * 17];   // [m][k], stride 17 -> conflict-free
  __shared__ float Bls[2][16 * 128];  // [k][n]

  const int t = threadIdx.x;
  const int wave = t >> 5, lane = t & 31;
  const int half = lane >> 4, l = lane & 15;
  const int wm = wave & 1, wn = wave >> 1;
  const int m_blk = blockIdx.x * 64;
  const int n_blk = blockIdx.y * 128;
  const int kk = t & 15, r = t >> 4;   // A staging coords (16 threads/row)
  const int nn = t & 127;              // B staging col (b32 fallback)

  v8f acc[2][2] = {};
  const int nstages = (K + 15) / 16;

#if HAS_ASYNC_LDS
  // ---- async staging: fire-and-forget global->LDS, tracked by ASYNCcnt ----
  auto stage = [&](int buf, int k0) {
#pragma unroll
    for (int i = 0; i < 4; ++i) {  // A tile 64x16, b32 (stride-17 rows)
      int m = r + i * 16, gk = k0 + kk;
      if (gk < K) async_g2l_b32(&A[(size_t)(m_blk + m) * K + gk], &Als[buf][m * 17 + kk]);
      else        Als[buf][m * 17 + kk] = 0.f;
    }
#pragma unroll
    for (int i = 0; i < 2; ++i) {  // B tile 16x128 as 512 aligned b128 chunks
      int c = t + i * 256, kr2 = c >> 5, n4 = (c & 31) * 4;
      int gk = k0 + kr2;
      if (gk < K) {
        async_g2l_b128(&B[(size_t)gk * N + n_blk + n4], &Bls[buf][kr2 * 128 + n4]);
      } else {
        Bls[buf][kr2 * 128 + n4 + 0] = 0.f; Bls[buf][kr2 * 128 + n4 + 1] = 0.f;
        Bls[buf][kr2 * 128 + n4 + 2] = 0.f; Bls[buf][kr2 * 128 + n4 + 3] = 0.f;
      }
    }
  };
#else
  // ---- fallback: register-prefetch double buffering ----
  float aReg[4], bReg[8];
  auto gload = [&](int k0) {
#pragma unroll
    for (int i = 0; i < 4; ++i) {
      int m = r + i * 16, gk = k0 + kk;
      aReg[i] = (gk < K) ? A[(size_t)(m_blk + m) * K + gk] : 0.f;
    }
#pragma unroll
    for (int i = 0; i < 8; ++i) {
      int kk2 = (t >> 7) + i * 2, gk = k0 + kk2;
      bReg[i] = (gk < K) ? B[(size_t)gk * N + n_blk + nn] : 0.f;
    }
  };
  auto lstore = [&](int buf) {
#pragma unroll
    for (int i = 0; i < 4; ++i) Als[buf][(r + i * 16) * 17 + kk] = aReg[i];
#pragma unroll
    for (int i = 0; i < 8; ++i) Bls[buf][((t >> 7) + i * 2) * 128 + nn] = bReg[i];
  };
#endif

  auto compute = [&](int buf) {
#pragma unroll
    for (int ks = 0; ks < 16; ks += 4) {
      v2f af[2], bf[2];
#pragma unroll
      for (int i = 0; i < 2; ++i) {  // A frag: M=l, K=ks+2*half{,+1}
        int m = wm * 32 + i * 16 + l;
        af[i].x = Als[buf][m * 17 + ks + 2 * half];
        af[i].y = Als[buf][m * 17 + ks + 2 * half + 1];
      }
#pragma unroll
      for (int j = 0; j < 2; ++j) {  // B frag: N=l, K=ks+2*half{,+1}
        int n = wn * 32 + j * 16 + l;
        bf[j].x = Bls[buf][(ks + 2 * half) * 128 + n];
        bf[j].y = Bls[buf][(ks + 2 * half + 1) * 128 + n];
      }
#pragma unroll
      for (int i = 0; i < 2; ++i)
#pragma unroll
        for (int j = 0; j < 2; ++j) acc[i][j] = WMMA_F32(af[i], bf[j], acc[i][j]);
    }
  };

#if HAS_ASYNC_LDS
  stage(0, 0);
  wait_async0();
  __syncthreads();
  for (int s = 0; s < nstages; ++s) {
    int cur = s & 1;
    if (s + 1 < nstages) stage(cur ^ 1, (s + 1) * 16);  // overlap with WMMAs
    compute(cur);
    wait_async0();
    __syncthreads();
  }
#else
  gload(0);
  lstore(0);
  __syncthreads();
  for (int s = 0; s < nstages; ++s) {
    int cur = s & 1;
    if (s + 1 < nstages) gload((s + 1) * 16);           // overlap with WMMAs
    compute(cur);
    if (s + 1 < nstages) lstore(cur ^ 1);
    __syncthreads();
  }
#endif

#pragma unroll
  for (int i = 0; i < 2; ++i)
#pragma unroll
    for (int j = 0; j < 2; ++j) {
      int ncol = n_blk + wn * 32 + j * 16 + l;
      float bv = bias[ncol];
#pragma unroll
      for (int v = 0; v < 8; ++v) {
        int row = m_blk + wm * 32 + i * 16 + v + 8 * half;
        float x = acc[i][j][v] + bv;
        if (ACT == 1) x = fmaxf(x, 0.f);
        C[(size_t)row * N + ncol] = x;
      }
    }
}

// ---------------------------------------------------------------------------
// Reparameterize: mean/log_var -> d_out, z_e = mean + exp(0.5 lv) * eps
// ---------------------------------------------------------------------------
__global__ void reparam_kernel(const float* __restrict__ enc,
                               const float* __restrict__ eps,
                               float* __restrict__ mean_out,
                               float* __restrict__ logvar_out,
                               float* __restrict__ z_e) {
  int i = blockIdx.x * blockDim.x + threadIdx.x;
  if (i >= Mrows * LAT) return;
  int row = i >> 6, d = i & 63;
  float m = enc[(size_t)row * 128 + d];
  float lv = enc[(size_t)row * 128 + 64 + d];
  mean_out[i] = m;
  logvar_out[i] = lv;
  z_e[i] = fmaf(__expf(0.5f * lv), eps[i], m);
}

// ---------------------------------------------------------------------------
// Codebook squared norms
// ---------------------------------------------------------------------------
__global__ void csq_kernel(const float* __restrict__ codebook, float* __restrict__ c_sq) {
  int n = blockIdx.x * blockDim.x + threadIdx.x;
  if (n >= KCODE) return;
  float s = 0.f;
#pragma unroll 8
  for (int k = 0; k < LAT; ++k) {
    float v = codebook[(size_t)n * LAT + k];
    s = fmaf(v, v, s);
  }
  c_sq[n] = s;
}

// ---------------------------------------------------------------------------
// VQ argmin: one block per 16 rows; WMMA cross terms vs all 1024 codes.
// Packed (dist_bits<<32 | idx) u64 LDS atomicMin -> first-min tie-break.
// ---------------------------------------------------------------------------
__global__ __launch_bounds__(256) void vq_argmin_kernel(
    const float* __restrict__ z_e, const float* __restrict__ codebook,
    const float* __restrict__ c_sq, int* __restrict__ indices) {
  __shared__ float Zls[16 * 65];  // [m][k], stride 65 -> conflict-free
  __shared__ float zsq[16];
  __shared__ unsigned long long best[16];

  const int t = threadIdx.x;
  const int row0 = blockIdx.x * 16;

  for (int i = t; i < 16 * LAT; i += 256) {
    int m = i >> 6, k = i & 63;
    Zls[m * 65 + k] = z_e[(size_t)(row0 + m) * LAT + k];
  }
  if (t < 16) best[t] = ~0ull;
  __syncthreads();
  if (t < 16) {
    float s = 0.f;
#pragma unroll 8
    for (int k = 0; k < LAT; ++k) s = fmaf(Zls[t * 65 + k], Zls[t * 65 + k], s);
    zsq[t] = s;
  }
  __syncthreads();

  const int wave = t >> 5, lane = t & 31, half = lane >> 4, l = lane & 15;
#pragma unroll
  for (int j = 0; j < 8; ++j) {  // 8 N-tiles of 16 codes per wave
    int n0 = (wave * 8 + j) * 16;
    v8f c = {};
#pragma unroll
    for (int ks = 0; ks < LAT; ks += 4) {
      v2f a, b;
      a.x = Zls[l * 65 + ks + 2 * half];
      a.y = Zls[l * 65 + ks + 2 * half + 1];
      // B[k][n] = codebook[n][k]  (256 KB codebook stays L2-resident)
      b.x = codebook[(size_t)(n0 + l) * LAT + ks + 2 * half];
      b.y = codebook[(size_t)(n0 + l) * LAT + ks + 2 * half + 1];
      c = WMMA_F32(a, b, c);
    }
#pragma unroll
    for (int v = 0; v < 8; ++v) {
      int row = v + 8 * half, ncol = n0 + l;
      float dist = fmaxf(zsq[row] - 2.f * c[v] + c_sq[ncol], 0.f);
      unsigned long long key =
          ((unsigned long long)__float_as_uint(dist) << 32) | (unsigned)ncol;
      atomicMin(&best[row], key);
    }
  }
  __syncthreads();
  if (t < 16) indices[row0 + t] = (int)(best[t] & 0xffffffffu);
}

// ---------------------------------------------------------------------------
// Zero scratch accumulators (d1 accumulator, d1, loss scalar)
// ---------------------------------------------------------------------------
__global__ void zero_kernel(float* __restrict__ p, int n) {
  int i = blockIdx.x * blockDim.x + threadIdx.x;
  if (i < n) p[i] = 0.f;
}

// ---------------------------------------------------------------------------
// Gather z_q + vq_loss partial sums
// ---------------------------------------------------------------------------
__global__ __launch_bounds__(256) void gather_loss_kernel(
    const float* __restrict__ z_e, const float* __restrict__ codebook,
    const int* __restrict__ idx, float* __restrict__ z_q,
    float* __restrict__ loss_sum) {
  __shared__ float red[256];
  int i = blockIdx.x * blockDim.x + threadIdx.x;
  float d2 = 0.f;
  if (i < Mrows * LAT) {
    int row = i >> 6, d = i & 63;
    float q = codebook[(size_t)idx[row] * LAT + d];
    z_q[i] = q;
    float df = z_e[i] - q;
    d2 = df * df;
  }
  red[threadIdx.x] = d2;
  __syncthreads();
  for (int s = 128; s > 0; s >>= 1) {
    if (threadIdx.x < s) red[threadIdx.x] += red[threadIdx.x + s];
    __syncthreads();
  }
  if (threadIdx.x == 0) atomicAdd(loss_sum, red[0]);
}

// ---------------------------------------------------------------------------
// D1 split-K (M=16, bandwidth-bound on W_d1): coalesced weight stream,
// 16 FMAs per weight against broadcast z_flat, fp32 atomic accumulate.
// ---------------------------------------------------------------------------
__global__ __launch_bounds__(256) void d1_splitk_kernel(
    const float* __restrict__ z_flat, const float* __restrict__ W,
    float* __restrict__ acc) {
  const int t = threadIdx.x;
  const int n = blockIdx.x * 64 + (t & 63);
  const int kpart = t >> 6;                      // 4 K slices per block
  const int k0 = blockIdx.y * 576 + kpart * 144; // grid.y = 64
  float s[Bdim] = {};
  for (int k = k0; k < k0 + 144; ++k) {
    float w = W[(size_t)k * 256 + n];
#pragma unroll
    for (int b = 0; b < Bdim; ++b) s[b] = fmaf(z_flat[(size_t)b * SD + k], w, s[b]);
  }
#pragma unroll
  for (int b = 0; b < Bdim; ++b) atomicAdd(&acc[b * 256 + n], s[b]);
}

__global__ void d1_epilogue_kernel(const float* __restrict__ acc,
                                   const float* __restrict__ bias,
                                   float* __restrict__ d1) {
  int i = blockIdx.x * blockDim.x + threadIdx.x;
  if (i >= Bdim * 256) return;
  d1[i] = fmaxf(acc[i] + bias[i & 255], 0.f);
}

// ---------------------------------------------------------------------------
// D2 (M=16, K=256, N=403200): d1 staged transposed in LDS; each wave streams
// one 16-column slab of W_d2 (coalesced 64B x 2 rows / frag), softplus out.
// Pure HBM-bandwidth kernel on the 413 MB W_d2 stream.
// ---------------------------------------------------------------------------
__global__ __launch_bounds__(256) void dec2_wmma_kernel(
    const float* __restrict__ d1, const float* __restrict__ W,
    const float* __restrict__ bias, float* __restrict__ out) {
  __shared__ float AT[256 * 17];  // [k][m], stride 17
  const int t = threadIdx.x;
  for (int i = t; i < 16 * 256; i += 256) {
    int m = i >> 8, k = i & 255;                 // coalesced read of d1 rows
    AT[k * 17 + m] = d1[m * 256 + k];
  }
  __syncthreads();

  const int wave = t >> 5, lane = t & 31, half = lane >> 4, l = lane & 15;
  const int n0 = (blockIdx.x * 8 + wave) * 16;
  if (n0 >= SF) return;

  v8f c = {};
#pragma unroll 8
  for (int ks = 0; ks < 256; ks += 4) {
    v2f a, b;
    a.x = AT[(ks + 2 * half) * 17 + l];
    a.y = AT[(ks + 2 * half + 1) * 17 + l];
    b.x = W[(size_t)(ks + 2 * half) * SF + n0 + l];
    b.y = W[(size_t)(ks + 2 * half + 1) * SF + n0 + l];
    c = WMMA_F32(a, b, c);
  }
#pragma unroll
  for (int v = 0; v < 8; ++v) {
    int row = v + 8 * half, ncol = n0 + l;
    float x = c[v] + bias[ncol];
    float sp = (x > 20.f) ? x : log1pf(__expf(x));  // stable softplus
    out[(size_t)row * SF + ncol] = sp;
  }
}

__global__ void loss_finalize_kernel(const float* __restrict__ loss_sum,
                                     float* __restrict__ loss_out) {
  loss_out[0] = loss_sum[0] * (1.0f / (float)(Mrows * LAT));
}

// ---------------------------------------------------------------------------
extern "C" void kernel_launch(void* const* d_in, const int* in_sizes, int n_in,
                              void* d_out, int out_size, void* d_ws, size_t ws_size,
                              hipStream_t stream) {
  const float* x    = (const float*)d_in[0];
  const float* eps  = (const float*)d_in[1];
  const float* W_e1 = (const float*)d_in[2];
  const float* b_e1 = (const float*)d_in[3];
  const float* W_e2 = (const float*)d_in[4];
  const float* b_e2 = (const float*)d_in[5];
  const float* W_e3 = (const float*)d_in[6];
  const float* b_e3 = (const float*)d_in[7];
  const float* codebook = (const float*)d_in[8];
  const float* W_d1 = (const float*)d_in[9];
  const float* b_d1 = (const float*)d_in[10];
  const float* W_d2 = (const float*)d_in[11];
  const float* b_d2 = (const float*)d_in[12];

  float* outp       = (float*)d_out;
  float* recon_out  = outp;                       // 16*576*700
  float* mean_out   = outp + (size_t)Bdim * SF;   // 589824
  float* logvar_out = mean_out + (size_t)Mrows * LAT;
  float* loss_out   = logvar_out + (size_t)Mrows * LAT;

  float* ws = (float*)d_ws;
  float* h1    = ws;                               // 9216*512
  float* h2    = ws + (size_t)Mrows * 512;         // 9216*256
  float* enc   = ws;                               // aliases h1 (dead by E3)
  float* z_e   = h2 + (size_t)Mrows * 256;         // 9216*64
  float* z_q   = z_e + (size_t)Mrows * LAT;        // 9216*64
  int*   idx   = (int*)(z_q + (size_t)Mrows * LAT);// 9216
  float* c_sq  = (float*)(idx + Mrows);            // 1024
  float* d1acc = c_sq + KCODE;                     // 4096
  float* d1    = d1acc + Bdim * 256;               // 4096
  float* lsum  = d1 + Bdim * 256;                  // 1

  // ---- encoder ----
  gemm_wmma<1><<<dim3(Mrows / 64, 512 / 128), 256, 0, stream>>>(x,  W_e1, b_e1, h1, Mrows, 512, Fdim);
  gemm_wmma<1><<<dim3(Mrows / 64, 256 / 128), 256, 0, stream>>>(h1, W_e2, b_e2, h2, Mrows, 256, 512);
  gemm_wmma<0><<<dim3(Mrows / 64, 128 / 128), 256, 0, stream>>>(h2, W_e3, b_e3, enc, Mrows, 128, 256);
  reparam_kernel<<<(Mrows * LAT) / 256, 256, 0, stream>>>(enc, eps, mean_out, logvar_out, z_e);

  // ---- vector quantization ----
  csq_kernel<<<KCODE / 256, 256, 0, stream>>>(codebook, c_sq);
  vq_argmin_kernel<<<Mrows / 16, 256, 0, stream>>>(z_e, codebook, c_sq, idx);
  zero_kernel<<<17, 256, 0, stream>>>(d1acc, Bdim * 256 * 2 + 1);  // d1acc, d1, lsum
  gather_loss_kernel<<<(Mrows * LAT) / 256, 256, 0, stream>>>(z_e, codebook, idx, z_q, lsum);

  // ---- decoder ----
  d1_splitk_kernel<<<dim3(4, 64), 256, 0, stream>>>(z_q, W_d1, d1acc);
  d1_epilogue_kernel<<<16, 256, 0, stream>>>(d1acc, b_d1, d1);
  dec2_wmma_kernel<<<SF / (16 * 8), 256, 0, stream>>>(d1, W_d2, b_d2, recon_out);
  loss_finalize_kernel<<<1, 1, 0, stream>>>(lsum, loss_out);
}